// trajectory_Distance_Loss_85177791414540
// MI455X (gfx1250) — compile-verified
//
#include <hip/hip_runtime.h>
#include <hip/hip_bf16.h>
#include <math.h>

typedef __attribute__((ext_vector_type(2))) float v2f;
typedef __attribute__((ext_vector_type(8))) float v8f;

namespace {
constexpr int   kB            = 256;
constexpr int   kK            = 11;
constexpr int   kD            = 128;
constexpr int   kRows         = kB * kK;          // 2816
constexpr int   kTilesPerSide = kRows / 16;       // 176 (exact)
constexpr int   kTotalTiles   = kTilesPerSide * 2;// 352
constexpr int   kWaves        = 8;                // waves per block
constexpr int   kBlocks       = kTotalTiles / kWaves; // 44
constexpr float kEps          = 1e-6f;
}

// One wave per 16-row tile. Uses V_WMMA_F32_16X16X4_F32 to compute the Gram
// matrix of the difference rows; its diagonal is the per-row squared L2 norm.
__global__ __launch_bounds__(256) void traj_loss_wmma_kernel(
    const float* __restrict__ nearT, const float* __restrict__ farT,
    const int*   __restrict__ naIdx, const int*   __restrict__ faIdx,
    const int*   __restrict__ neIdx, const int*   __restrict__ feIdx,
    const float* __restrict__ nTgt,  const float* __restrict__ fTgt,
    float* __restrict__ partial)
{
    __shared__ float ldsC[kWaves][8][32];  // C-matrix dump: [wave][vgpr][lane]
    __shared__ float red[256];

    const int tid  = threadIdx.x;
    const int wave = tid >> 5;
    const int lane = tid & 31;
    const int tile = blockIdx.x * kWaves + wave;   // 0..351, uniform per wave

    const bool  isFar = (tile >= kTilesPerSide);
    const int   t     = isFar ? (tile - kTilesPerSide) : tile;
    const float* src  = isFar ? farT  : nearT;
    const int*  idxA  = isFar ? faIdx : naIdx;
    const int*  idxE  = isFar ? feIdx : neIdx;
    const float* tgt  = isFar ? fTgt  : nTgt;

    // A-matrix 16x4 f32 layout (wave32): lane = h*16 + m holds row m,
    // K-elements {2h, 2h+1} of each K=4 chunk.
    const int m = lane & 15;            // row within tile this lane feeds
    const int h = lane >> 4;            // which half of the K=4 chunk
    const int r = t * 16 + m;           // global row 0..2815
    const int b = r / kK;               // gather column: src[idx, b]

    const long baseA = ((long)idxA[r] * kB + b) * kD;
    const long baseE = ((long)idxE[r] * kB + b) * kD;
    const v2f* __restrict__ pA = (const v2f*)(src + baseA);  // 8B-aligned
    const v2f* __restrict__ pE = (const v2f*)(src + baseE);

    v8f c = {};
    #pragma unroll
    for (int cc = 0; cc < kD / 4; ++cc) {
        // chunk cc covers K-elements 4cc..4cc+3 of row m; this lane loads
        // elements 4cc+2h, 4cc+2h+1  -> float2 index 2cc + h.
        v2f va = pA[2 * cc + h];
        v2f ve = pE[2 * cc + h];
        v2f d;
        d.x = va.x - ve.x + kEps;
        d.y = va.y - ve.y + kEps;
        // Same registers as A and B == A * A^T (B layout is A-layout transposed).
        c = __builtin_amdgcn_wmma_f32_16x16x4_f32(
                /*neg_a=*/false, d, /*neg_b=*/false, d,
                /*c_mod=*/(short)0, c, /*reuse_a=*/false, /*reuse_b=*/false);
    }

    // Spill C to LDS so we can pick the diagonal without dynamic VGPR indexing.
    #pragma unroll
    for (int g = 0; g < 8; ++g) ldsC[wave][g][lane] = c[g];
    __syncthreads();

    float val = 0.0f;
    if (lane < 16) {
        const int mm = lane;                            // tile row 0..15
        // Diagonal D[i][i]: i<8 -> (lane i, vgpr i); i>=8 -> (lane 16+i, vgpr i-8)
        const int srcLane = (mm < 8) ? mm : (16 + mm);
        const int g       = mm & 7;
        const float ss    = ldsC[wave][g][srcLane];     // ||a-e+eps||^2
        const float dist  = sqrtf(ss);
        const float loss  = expf(-dist) - tgt[t * 16 + mm];
        val = loss * loss;
    }

    // Block reduction -> one partial per block.
    red[tid] = val;
    __syncthreads();
    #pragma unroll
    for (int s = 128; s > 0; s >>= 1) {
        if (tid < s) red[tid] += red[tid + s];
        __syncthreads();
    }
    if (tid == 0) partial[blockIdx.x] = red[0];
}

__global__ __launch_bounds__(64) void traj_loss_reduce_kernel(
    const float* __restrict__ partial, float* __restrict__ out, int n)
{
    __shared__ float red[64];
    const int tid = threadIdx.x;
    float v = 0.0f;
    for (int i = tid; i < n; i += 64) v += partial[i];
    red[tid] = v;
    __syncthreads();
    #pragma unroll
    for (int s = 32; s > 0; s >>= 1) {
        if (tid < s) red[tid] += red[tid + s];
        __syncthreads();
    }
    if (tid == 0) out[0] = red[0];
}

extern "C" void kernel_launch(void* const* d_in, const int* in_sizes, int n_in,
                              void* d_out, int out_size, void* d_ws, size_t ws_size,
                              hipStream_t stream) {
    const float* nearT = (const float*)d_in[0];
    const float* farT  = (const float*)d_in[1];
    const int*   naIdx = (const int*)d_in[2];
    const int*   faIdx = (const int*)d_in[3];
    const int*   neIdx = (const int*)d_in[4];
    const int*   feIdx = (const int*)d_in[5];
    const float* nTgt  = (const float*)d_in[6];
    const float* fTgt  = (const float*)d_in[7];

    float* partial = (float*)d_ws;  // kBlocks floats of scratch

    traj_loss_wmma_kernel<<<kBlocks, 256, 0, stream>>>(
        nearT, farT, naIdx, faIdx, neIdx, feIdx, nTgt, fTgt, partial);
    traj_loss_reduce_kernel<<<1, 64, 0, stream>>>(partial, (float*)d_out, kBlocks);
}